// GraphIsomorphismNetworkLayer_88742614270404
// MI455X (gfx1250) — compile-verified
//
#include <hip/hip_runtime.h>

// ---------------------------------------------------------------------------
// GIN layer on MI455X (gfx1250, wave32).
// Phase 1: h = x @ W^T + b  via V_WMMA_F32_16X16X4_F32 (fp32 matrix path).
// Phase 2: out = relu(h + sum_k h[a2a[:,k]])  -- bandwidth-bound row gather.
// ---------------------------------------------------------------------------

typedef __attribute__((ext_vector_type(2))) float v2f;
typedef __attribute__((ext_vector_type(8))) float v8f;

#define D_FEAT 256
#define MAX_NB 16
// Padded LDS row stride (floats): 260 -> row offset 1040B (16B aligned),
// bank(row,k) = (4*row + k) mod 64 -> conflict-free across the 16 rows.
#define LDS_STRIDE 260

// ---------------------------------------------------------------------------
// Phase 1: one block per 16-atom M tile; 8 waves, each wave owns two 16-col
// N tiles. K loop in steps of 4 (64 WMMAs per tile).
// A frag (16x4 f32): lane<16 -> (M=lane,   K=k0..k0+1)
//                    lane>=16 -> (M=lane-16, K=k0+2..k0+3)   (ISA 7.12.2)
// B frag (4x16 f32): same split with N=lane&15 striped across lanes.
// C/D (16x16 f32, 8 VGPRs): elem e, lane -> M = e + 8*(lane>=16), N = lane&15.
// ---------------------------------------------------------------------------
__global__ __launch_bounds__(256) void gin_gemm_wmma(
    const float* __restrict__ x, const float* __restrict__ W,
    const float* __restrict__ bias, float* __restrict__ h, int N)
{
    __shared__ float sx[16 * LDS_STRIDE];
    const int row0 = blockIdx.x * 16;
    const int tid  = threadIdx.x;

    // Cooperative, coalesced load of the 16x256 x-tile into padded LDS.
    // 1024 float4 elements / 256 threads = 4 per thread.
    for (int e4 = tid; e4 < 16 * (D_FEAT / 4); e4 += 256) {
        const int r  = e4 >> 6;     // row 0..15
        const int c4 = e4 & 63;     // float4 column
        const int row = row0 + r;
        float4 v = make_float4(0.f, 0.f, 0.f, 0.f);
        if (row < N) v = ((const float4*)(x + (size_t)row * D_FEAT))[c4];
        *(float4*)(&sx[r * LDS_STRIDE + c4 * 4]) = v;
    }
    __syncthreads();

    const int lane = tid & 31;
    const int wave = tid >> 5;
    const int half = lane >> 4;   // 0 or 1 (selects K pair)
    const int lr   = lane & 15;   // M (for A) / N (for B, C, D)

    const float* ap = &sx[lr * LDS_STRIDE + 2 * half];

    for (int nt = wave * 2; nt < wave * 2 + 2; ++nt) {
        const int jbase = nt * 16;
        // W is [D_OUT, D_IN] row-major: W[j][k] -> exactly the B layout.
        const float* wp = W + (size_t)(jbase + lr) * D_FEAT + 2 * half;

        v8f acc = {};
#pragma unroll 8
        for (int k0 = 0; k0 < D_FEAT; k0 += 4) {
            v2f a = *(const v2f*)(ap + k0);   // ds_load_b64, conflict-free
            v2f b = *(const v2f*)(wp + k0);   // global_load_b64, L2-resident W
            acc = __builtin_amdgcn_wmma_f32_16x16x4_f32(
                /*neg_a=*/false, a, /*neg_b=*/false, b,
                /*c_mod=*/(short)0, acc, /*reuse_a=*/false, /*reuse_b=*/false);
        }

        const float bj = bias[jbase + lr];
#pragma unroll
        for (int e = 0; e < 8; ++e) {
            const int m    = e + half * 8;
            const int atom = row0 + m;
            if (atom < N) {
                float v = acc[e] + bj;
                if (atom == 0) v = 0.0f;     // h row 0 is padding -> zero
                h[(size_t)atom * D_FEAT + jbase + lr] = v;
            }
        }
    }
}

// ---------------------------------------------------------------------------
// Phase 2: out[i] = relu(h[i] + sum_{k<16} h[b_from_a[a_from_b[i,k]]]).
// 64 threads per atom (one float4 each), 4 atoms per 256-thread block.
// Indices resolved once into LDS; row gathers are fully coalesced 1KB reads
// that should largely hit L2 (h = 51MB << 192MB L2).
// ---------------------------------------------------------------------------
__global__ __launch_bounds__(256) void gin_aggregate(
    const float* __restrict__ h, const int* __restrict__ b_from_a,
    const int* __restrict__ a_from_b, float* __restrict__ out, int N)
{
    __shared__ int sidx[4 * MAX_NB];
    const int al   = threadIdx.x >> 6;   // atom slot in block (0..3)
    const int t    = threadIdx.x & 63;   // float4 lane within row
    const int atom = blockIdx.x * 4 + al;

    if (atom < N && t < MAX_NB) {
        const int bond = a_from_b[(size_t)atom * MAX_NB + t];
        sidx[al * MAX_NB + t] = b_from_a[bond];
    }
    __syncthreads();
    if (atom >= N) return;

    const float4* hp = (const float4*)h;
    float4 acc = hp[(size_t)atom * (D_FEAT / 4) + t];

#pragma unroll
    for (int k = 0; k < MAX_NB; ++k) {
        const int src = sidx[al * MAX_NB + k];
        const float4 m = hp[(size_t)src * (D_FEAT / 4) + t];
        acc.x += m.x; acc.y += m.y; acc.z += m.z; acc.w += m.w;
    }

    acc.x = fmaxf(acc.x, 0.0f);
    acc.y = fmaxf(acc.y, 0.0f);
    acc.z = fmaxf(acc.z, 0.0f);
    acc.w = fmaxf(acc.w, 0.0f);

    ((float4*)out)[(size_t)atom * (D_FEAT / 4) + t] = acc;
}

// ---------------------------------------------------------------------------
// Launch: inputs in setup_inputs() order:
//   d_in[0] = x        [N, 256]  f32
//   d_in[1] = W        [256,256] f32
//   d_in[2] = b        [256]     f32
//   d_in[3] = b_from_a [400000]  i32
//   d_in[4] = a_from_b [N, 16]   i32
// d_ws holds h [N, 256] f32 (51.2 MB).
// ---------------------------------------------------------------------------
extern "C" void kernel_launch(void* const* d_in, const int* in_sizes, int n_in,
                              void* d_out, int out_size, void* d_ws, size_t ws_size,
                              hipStream_t stream)
{
    const float* x        = (const float*)d_in[0];
    const float* W        = (const float*)d_in[1];
    const float* b        = (const float*)d_in[2];
    const int*   b_from_a = (const int*)d_in[3];
    const int*   a_from_b = (const int*)d_in[4];
    float*       out      = (float*)d_out;
    float*       h        = (float*)d_ws;

    const int N = in_sizes[0] / D_FEAT;   // 50000

    const int mblocks = (N + 15) / 16;    // 3125 M tiles
    gin_gemm_wmma<<<mblocks, 256, 0, stream>>>(x, W, b, h, N);

    const int ablocks = (N + 3) / 4;      // 12500 aggregation blocks
    gin_aggregate<<<ablocks, 256, 0, stream>>>(h, b_from_a, a_from_b, out, N);
}